// LayerNorm_37855841747141
// MI455X (gfx1250) — compile-verified
//
#include <hip/hip_runtime.h>

// LayerNorm over last dim D=1024 of a (4,2048,1024) fp32 tensor, per-position affine.
// Memory-bound: 128 MiB traffic -> ~5.8us at 23.3 TB/s. One workgroup (256 thr = 8
// wave32) per row; float4 per thread. Gamma/beta are fetched via the CDNA5 async
// global->LDS path (ASYNCcnt) so their latency hides under the two reduction passes.

typedef float v4f __attribute__((ext_vector_type(4)));
typedef int   v4i __attribute__((ext_vector_type(4)));

#define LN_D    1024
#define LN_TPB  256           // 8 waves of 32
#define LN_EPS  5e-6f

#if defined(__gfx1250__) && __has_builtin(__builtin_amdgcn_global_load_async_to_lds_b128)
#define LN_HAS_ASYNC 1
#else
#define LN_HAS_ASYNC 0
#endif

#if LN_HAS_ASYNC
// Builtin signature (from clang diagnostics): (v4i AS1*, v4i AS3*, imm off, imm cpol)
typedef __attribute__((address_space(1))) v4i* ln_gptr;
typedef __attribute__((address_space(3))) v4i* ln_lptr;
#endif

__device__ __forceinline__ void ln_copy16_to_lds(const float* g, float* l) {
#if LN_HAS_ASYNC
    __builtin_amdgcn_global_load_async_to_lds_b128(
        (ln_gptr)const_cast<float*>(g), (ln_lptr)l, 0, 0);
#else
    *(v4f*)l = __builtin_nontemporal_load((const v4f*)g);
#endif
}

__global__ __launch_bounds__(LN_TPB) void LayerNorm_37855841747141_kernel(
    const float* __restrict__ in, const float* __restrict__ gamma,
    const float* __restrict__ beta, float* __restrict__ out) {
    __shared__ float s_gamma[LN_D];
    __shared__ float s_beta[LN_D];
    __shared__ float s_red[16];   // [0..7] pass-1 wave sums, [8..15] pass-2 wave sums

    const int  tid  = threadIdx.x;
    const int  lane = tid & 31;
    const int  wave = tid >> 5;
    const long base = (long)blockIdx.x * LN_D;
    const int  off  = tid * 4;

    // Kick off async gamma/beta -> LDS (overlaps with everything below).
    ln_copy16_to_lds(gamma + base + off, &s_gamma[off]);
    ln_copy16_to_lds(beta  + base + off, &s_beta[off]);

    // Streamed input, read once: nontemporal b128 load into registers.
    v4f x = __builtin_nontemporal_load((const v4f*)(in + base + off));

    // ---- Pass 1: mean ----
    float s = (x[0] + x[1]) + (x[2] + x[3]);
#pragma unroll
    for (int m = 16; m > 0; m >>= 1) s += __shfl_xor(s, m, 32);
    if (lane == 0) s_red[wave] = s;
    __syncthreads();
    float tot = 0.f;
#pragma unroll
    for (int w = 0; w < LN_TPB / 32; ++w) tot += s_red[w];
    const float mean = tot * (1.0f / LN_D);

    // ---- Pass 2: population variance from register-held x ----
    const float d0 = x[0] - mean, d1 = x[1] - mean, d2 = x[2] - mean, d3 = x[3] - mean;
    float q = (d0 * d0 + d1 * d1) + (d2 * d2 + d3 * d3);
#pragma unroll
    for (int m = 16; m > 0; m >>= 1) q += __shfl_xor(q, m, 32);
    if (lane == 0) s_red[8 + wave] = q;
    __syncthreads();
    float vtot = 0.f;
#pragma unroll
    for (int w = 0; w < LN_TPB / 32; ++w) vtot += s_red[8 + w];
    const float rstd = __frsqrt_rn(vtot * (1.0f / LN_D) + LN_EPS);

    // Drain the async gamma/beta transfers, then finish. The "memory" clobber also
    // orders the LDS reads below after the wait at the compiler level.
    asm volatile("s_wait_asynccnt 0" ::: "memory");
    const v4f g = *(const v4f*)&s_gamma[off];
    const v4f b = *(const v4f*)&s_beta[off];
    v4f o;
    o[0] = __fmaf_rn(d0 * rstd, g[0], b[0]);
    o[1] = __fmaf_rn(d1 * rstd, g[1], b[1]);
    o[2] = __fmaf_rn(d2 * rstd, g[2], b[2]);
    o[3] = __fmaf_rn(d3 * rstd, g[3], b[3]);
    __builtin_nontemporal_store(o, (v4f*)(out + base + off));
}

extern "C" void kernel_launch(void* const* d_in, const int* in_sizes, int n_in,
                              void* d_out, int out_size, void* d_ws, size_t ws_size,
                              hipStream_t stream) {
    (void)n_in; (void)out_size; (void)d_ws; (void)ws_size;
    const float* in    = (const float*)d_in[0];
    const float* gamma = (const float*)d_in[1];
    const float* beta  = (const float*)d_in[2];
    float*       out   = (float*)d_out;
    const int rows = in_sizes[0] / LN_D;   // B*S = 8192
    LayerNorm_37855841747141_kernel<<<dim3(rows), dim3(LN_TPB), 0, stream>>>(
        in, gamma, beta, out);
}